// MultiBoxLoss_69698729279762
// MI455X (gfx1250) — compile-verified
//
#include <hip/hip_runtime.h>

#define BB 64
#define PP 8732
#define TT 32
#define CC 81
#define TILE_P 128
#define NTILES ((PP + TILE_P - 1) / TILE_P)   // 69

typedef __attribute__((ext_vector_type(2))) float v2f;
typedef __attribute__((ext_vector_type(8))) float v8f;
typedef __attribute__((ext_vector_type(4))) unsigned int v4u;
typedef __attribute__((ext_vector_type(8))) int v8i;
typedef __attribute__((ext_vector_type(4))) int v4i;

__device__ __forceinline__ float smooth_l1(float d) {
    float ad = fabsf(d);
    return (ad < 1.0f) ? 0.5f * d * d : ad - 0.5f;
}

// ---------------------------------------------------------------------------
// K1: per-batch matching, encode, smooth-L1 loss, num_pos, conf_t
// ---------------------------------------------------------------------------
__global__ __launch_bounds__(256) void k_match(const float* __restrict__ loc,
                                               const float* __restrict__ priors,
                                               const float* __restrict__ tboxes,
                                               const int* __restrict__ tlabels,
                                               int* __restrict__ conf_t,
                                               float* __restrict__ lossl_b,
                                               int* __restrict__ npos_b) {
    int b = blockIdx.x;
    int tid = threadIdx.x;

    __shared__ float4 s_truth[TT];                 // corner form
    __shared__ int s_lab[TT];
    __shared__ unsigned long long s_best[TT];      // (ov_bits<<32)|(~p): max -> first-index argmax
    __shared__ float s_btov[PP];
    __shared__ unsigned short s_bti[PP];
    __shared__ float s_redf[256];
    __shared__ int s_redi[256];

    if (tid < TT) {
        const float* tb = tboxes + ((size_t)b * TT + tid) * 4;
        s_truth[tid] = make_float4(tb[0], tb[1], tb[2], tb[3]);
        s_lab[tid] = tlabels[b * TT + tid];
        s_best[tid] = 0xFFFFFFFFull;               // ov=0, p=0
    }
    __syncthreads();

    const float4* pr4 = reinterpret_cast<const float4*>(priors);
    // Pass B: per-prior best truth + per-truth best prior
    for (int p = tid; p < PP; p += 256) {
        float4 pr = pr4[p];                        // cx, cy, w, h
        float pxmin = pr.x - pr.z * 0.5f, pymin = pr.y - pr.w * 0.5f;
        float pxmax = pr.x + pr.z * 0.5f, pymax = pr.y + pr.w * 0.5f;
        float areaP = (pxmax - pxmin) * (pymax - pymin);
        float bestov = -1.0f;
        int bestt = 0;
        for (int t = 0; t < TT; ++t) {
            float4 tr = s_truth[t];
            float ix = fmaxf(fminf(pxmax, tr.z) - fmaxf(pxmin, tr.x), 0.0f);
            float iy = fmaxf(fminf(pymax, tr.w) - fmaxf(pymin, tr.y), 0.0f);
            float inter = ix * iy;
            float areaT = (tr.z - tr.x) * (tr.w - tr.y);
            float ov = inter / (areaT + areaP - inter);
            if (ov > bestov) { bestov = ov; bestt = t; }
            if (ov > 0.0f) {
                unsigned long long pk =
                    (((unsigned long long)__float_as_uint(ov)) << 32) |
                    (unsigned long long)(0xFFFFFFFFu - (unsigned)p);
                atomicMax(&s_best[t], pk);
            }
        }
        s_btov[p] = bestov;
        s_bti[p] = (unsigned short)bestt;
    }
    __syncthreads();

    // Pass C: force each truth's best prior (sequential: last j wins on duplicates)
    if (tid == 0) {
        for (int j = 0; j < TT; ++j) {
            unsigned p = 0xFFFFFFFFu - (unsigned)(s_best[j] & 0xFFFFFFFFull);
            s_btov[p] = 2.0f;
            s_bti[p] = (unsigned short)j;
        }
    }
    __syncthreads();

    // Pass D: conf_t, encode, smooth-L1
    const float4* loc4 = reinterpret_cast<const float4*>(loc);
    float ll = 0.0f;
    int np = 0;
    for (int p = tid; p < PP; p += 256) {
        float ov = s_btov[p];
        int t = (int)s_bti[p];
        int c = (ov < 0.5f) ? 0 : s_lab[t] + 1;
        conf_t[(size_t)b * PP + p] = c;
        if (c > 0) {
            np++;
            float4 pr = pr4[p];
            float4 tr = s_truth[t];
            float g0 = ((tr.x + tr.z) * 0.5f - pr.x) / (0.1f * pr.z);
            float g1 = ((tr.y + tr.w) * 0.5f - pr.y) / (0.1f * pr.w);
            float g2 = logf((tr.z - tr.x) / pr.z) * 5.0f;   // / 0.2
            float g3 = logf((tr.w - tr.y) / pr.w) * 5.0f;
            float4 l4 = loc4[(size_t)b * PP + p];
            ll += smooth_l1(l4.x - g0) + smooth_l1(l4.y - g1) +
                  smooth_l1(l4.z - g2) + smooth_l1(l4.w - g3);
        }
    }
    s_redf[tid] = ll;
    s_redi[tid] = np;
    __syncthreads();
    for (int s = 128; s > 0; s >>= 1) {
        if (tid < s) { s_redf[tid] += s_redf[tid + s]; s_redi[tid] += s_redi[tid + s]; }
        __syncthreads();
    }
    if (tid == 0) { lossl_b[b] = s_redf[0]; npos_b[b] = s_redi[0]; }
}

// ---------------------------------------------------------------------------
// K2: per-prior cross entropy. TDM stages the conf tile into LDS; WMMA f32
// 16x16x4 with B=ones does the 16-row sum-of-exp reductions on the matrix pipe.
// 81 classes = 20 full unguarded K=4 chunks (0..79) + scalar tail (class 80).
// Raw v_exp_f32/v_log_f32: args are (x - rowmax) <= 0; anything below the
// hardware denorm edge contributes 0 to the softmax sum.
// ---------------------------------------------------------------------------
__global__ __launch_bounds__(256) void k_ce(const float* __restrict__ conf,
                                            const int* __restrict__ conf_t,
                                            float* __restrict__ ce) {
    __shared__ float tile[TILE_P * CC];            // 41,472 B

    int b = blockIdx.x / NTILES;
    int tidx = blockIdx.x % NTILES;
    int p0 = tidx * TILE_P;
    int n = min(TILE_P, PP - p0);
    unsigned nelem = (unsigned)(n * CC);

    // Wave 0 issues the TDM descriptor and waits on TENSORcnt.
    if (threadIdx.x < 32) {
        unsigned long long gaddr =
            (unsigned long long)(const void*)(conf + ((size_t)b * PP + p0) * CC);
        unsigned ldsoff = (unsigned)(size_t)(void*)tile;
        v4u g0;
        g0[0] = 1u;                                           // count=1, user mode
        g0[1] = ldsoff;                                       // lds_addr
        g0[2] = (unsigned)gaddr;                              // global_addr lo
        g0[3] = (unsigned)((gaddr >> 32) & 0x1FFFFFFull) | (2u << 30); // addr hi | type=2
        v8i g1;
        g1[0] = (int)(2u << 16);                              // data_size = 4B
        g1[1] = (int)((nelem & 0xFFFFu) << 16);               // tensor_dim0[15:0]
        g1[2] = (int)(nelem >> 16);                           // tensor_dim0[31:16]
        g1[3] = (int)((nelem & 0xFFFFu) << 16);               // tile_dim0
        g1[4] = 0;                                            // tile_dim1/2 unused
        g1[5] = (int)nelem;                                   // tensor_dim0_stride lo
        g1[6] = 0;
        g1[7] = 0;
        v4i gz4 = {0, 0, 0, 0};
        v8i gz8 = {0, 0, 0, 0, 0, 0, 0, 0};
        __builtin_amdgcn_tensor_load_to_lds(g0, g1, gz4, gz4, gz8, 0);
        __builtin_amdgcn_s_wait_tensorcnt(0);
    }
    __syncthreads();

    int wave = threadIdx.x >> 5;
    int lane = threadIdx.x & 31;
    int row = lane & 15;          // matrix row (prior within wave's 16)
    int half = lane >> 4;         // A-matrix half-lane split: K pairs {0,1} vs {2,3}
    int rg = wave * 16 + row;     // row within tile [0,128)
    const float* rp = &tile[rg * CC];

    // Row max: per-lane over its K-residues (classes 0..79), tail class 80 on
    // half 0, then combine halves via lane-16 xor.
    float mx = -3.402823466e38f;
    for (int cb = 0; cb < 80; cb += 4) {
        int c0 = cb + half * 2;
        mx = fmaxf(mx, fmaxf(rp[c0], rp[c0 + 1]));
    }
    if (half == 0) mx = fmaxf(mx, rp[80]);
    mx = fmaxf(mx, __shfl_xor(mx, 16, 32));

    // Sum of exp via WMMA: A = 16x4 exp tile, B = ones -> C rows accumulate sums
    v8f acc = {0.f, 0.f, 0.f, 0.f, 0.f, 0.f, 0.f, 0.f};
    v2f ones = {1.0f, 1.0f};
    for (int cb = 0; cb < 80; cb += 4) {
        int c0 = cb + half * 2;
        v2f a;
        a.x = __builtin_amdgcn_exp2f((rp[c0] - mx) * 1.44269504089f);
        a.y = __builtin_amdgcn_exp2f((rp[c0 + 1] - mx) * 1.44269504089f);
        acc = __builtin_amdgcn_wmma_f32_16x16x4_f32(false, a, false, ones,
                                                    (short)0, acc, false, false);
    }

    // Extract row sums: M=0..7 -> acc[M] on lane 0; M=8..15 -> acc[M-8] on lane 16
    float rs = 0.0f;
#pragma unroll
    for (int i = 0; i < 8; ++i) {
        float lo = __shfl(acc[i], 0, 32);
        float hi = __shfl(acc[i], 16, 32);
        rs = (row == i) ? lo : rs;
        rs = (row == i + 8) ? hi : rs;
    }
    // Tail: class 80
    rs += __builtin_amdgcn_exp2f((rp[80] - mx) * 1.44269504089f);

    if (half == 0 && rg < n) {
        size_t p = (size_t)b * PP + (p0 + rg);
        int clsT = conf_t[p];
        float lse = mx + __builtin_amdgcn_logf(rs) * 0.69314718056f;  // log2 * ln2
        ce[p] = lse - rp[clsT];
    }
}

// ---------------------------------------------------------------------------
// K3: per-batch hard-negative mining via radix-select (K-th largest of mine),
// loss_c = sum_pos ce + sum_{mine>v*} mine + (K-G)*v*
// ---------------------------------------------------------------------------
__global__ __launch_bounds__(256) void k_mine(const float* __restrict__ ce,
                                              const int* __restrict__ conf_t,
                                              const int* __restrict__ npos_b,
                                              float* __restrict__ lossc_b) {
    int b = blockIdx.x;
    int tid = threadIdx.x;
    const float* ceb = ce + (size_t)b * PP;
    const int* ctb = conf_t + (size_t)b * PP;

    __shared__ unsigned hist[256];
    __shared__ float red[256];
    __shared__ unsigned sh_prefix;
    __shared__ int sh_remK;

    // sum of ce over positives
    float sp = 0.0f;
    for (int p = tid; p < PP; p += 256)
        if (ctb[p] > 0) sp += ceb[p];
    red[tid] = sp;
    __syncthreads();
    for (int s = 128; s > 0; s >>= 1) {
        if (tid < s) red[tid] += red[tid + s];
        __syncthreads();
    }
    float spce = red[0];
    __syncthreads();

    int npos = npos_b[b];
    int K = min(3 * npos, PP - 1);
    float lossc = spce;

    if (K > 0) {
        unsigned prefix = 0;
        int remK = K;
        for (int pass = 0; pass < 4; ++pass) {
            int shift = 24 - 8 * pass;
            hist[tid] = 0;
            __syncthreads();
            for (int p = tid; p < PP; p += 256) {
                unsigned u = (ctb[p] > 0) ? 0u : __float_as_uint(ceb[p]);
                if (pass == 0 || (u >> (shift + 8)) == prefix)
                    atomicAdd(&hist[(u >> shift) & 255u], 1u);
            }
            __syncthreads();
            if (tid == 0) {
                unsigned acc = 0;
                int bin = 0;
                for (int bnum = 255; bnum >= 0; --bnum) {
                    unsigned h = hist[bnum];
                    if (acc + h >= (unsigned)remK) { bin = bnum; break; }
                    acc += h;
                }
                sh_prefix = (prefix << 8) | (unsigned)bin;
                sh_remK = remK - (int)acc;   // K-slots left among == chosen-prefix elems
            }
            __syncthreads();
            prefix = sh_prefix;
            remK = sh_remK;
            __syncthreads();
        }
        float vstar = __uint_as_float(prefix);
        float sg = 0.0f;
        for (int p = tid; p < PP; p += 256) {
            unsigned u = (ctb[p] > 0) ? 0u : __float_as_uint(ceb[p]);
            if (u > prefix) sg += __uint_as_float(u);
        }
        red[tid] = sg;
        __syncthreads();
        for (int s = 128; s > 0; s >>= 1) {
            if (tid < s) red[tid] += red[tid + s];
            __syncthreads();
        }
        lossc += red[0] + (float)remK * vstar;   // ties all equal v*: stable-rank exact
    }
    if (tid == 0) lossc_b[b] = lossc;
}

// ---------------------------------------------------------------------------
// K4: final scalar reduction
// ---------------------------------------------------------------------------
__global__ void k_final(const float* __restrict__ lossl_b,
                        const float* __restrict__ lossc_b,
                        const int* __restrict__ npos_b,
                        float* __restrict__ out) {
    if (threadIdx.x == 0 && blockIdx.x == 0) {
        float L = 0.0f, C = 0.0f;
        long long N = 0;
        for (int b = 0; b < BB; ++b) {
            L += lossl_b[b];
            C += lossc_b[b];
            N += npos_b[b];
        }
        float Nf = fmaxf((float)N, 1.0f);
        out[0] = L / Nf;
        out[1] = C / Nf;
    }
}

extern "C" void kernel_launch(void* const* d_in, const int* in_sizes, int n_in,
                              void* d_out, int out_size, void* d_ws, size_t ws_size,
                              hipStream_t stream) {
    const float* loc = (const float*)d_in[0];
    const float* conf = (const float*)d_in[1];
    const float* priors = (const float*)d_in[2];
    const float* tboxes = (const float*)d_in[3];
    const int* tlabels = (const int*)d_in[4];
    float* out = (float*)d_out;

    const size_t BP = (size_t)BB * PP;
    char* ws = (char*)d_ws;
    int* conf_t = (int*)ws;                         // B*P ints
    float* ce = (float*)(ws + BP * 4);              // B*P floats
    float* lossl_b = (float*)(ws + 2 * BP * 4);     // B floats
    float* lossc_b = lossl_b + BB;                  // B floats
    int* npos_b = (int*)(lossc_b + BB);             // B ints

    k_match<<<BB, 256, 0, stream>>>(loc, priors, tboxes, tlabels, conf_t, lossl_b, npos_b);
    k_ce<<<BB * NTILES, 256, 0, stream>>>(conf, conf_t, ce);
    k_mine<<<BB, 256, 0, stream>>>(ce, conf_t, npos_b, lossc_b);
    k_final<<<1, 32, 0, stream>>>(lossl_b, lossc_b, npos_b, out);
}